// KAN_12936441496127
// MI455X (gfx1250) — compile-verified
//
#include <hip/hip_runtime.h>
#include <math.h>

#define BATCH     2048
#define IN_COUNT  256
#define OUT_COUNT 256
#define N_SPLINES 5
#define TAB_ELEMS (IN_COUNT * OUT_COUNT)

#define LN2_F     0.69314718055994531f
#define LOG2E_F   1.44269504088896340f

typedef __attribute__((ext_vector_type(2))) float v2f;
typedef __attribute__((ext_vector_type(8))) float v8f;

// Raw gfx1250 transcendental ops: v_exp_f32 / v_log_f32 (both base-2).
__device__ __forceinline__ float fast_exp2(float v) { return __builtin_amdgcn_exp2f(v); }
__device__ __forceinline__ float fast_log2(float v) { return __builtin_amdgcn_logf(v); }

__device__ __forceinline__ float softplus_precise(float v) {
    return fmaxf(v, 0.0f) + log1pf(__expf(-fabsf(v)));
}

// ---------------------------------------------------------------------------
// Kernel 1: evaluate the 5 splines at w_norm[o,i] once per call.
// Tables are stored transposed [i][o] and pre-folded into exp2/log2 space:
//   T4.x = b1 * gamma/OUT * ln2   (outer log1p -> log2, scale folded here)
//   T4.y = b2 * ln2               (inner log1p -> log2, scale folded here)
//   T4.z = b3 * log2(e)           (exp -> exp2)
//   T4.w = b4                     (pow exponent, stays in log2 space)
//   B5   = b5 * gamma/OUT         (linear term, consumed by WMMA)
// ---------------------------------------------------------------------------
__global__ void kan_spline_precompute(const float* __restrict__ w,
                                      const float* __restrict__ raw_gamma,
                                      const float* __restrict__ breaks,   // [5,17]
                                      const float* __restrict__ coefs,    // [5,16,4]
                                      const float* __restrict__ mu_p,
                                      const float* __restrict__ sigma_p,
                                      float4* __restrict__ tab4,
                                      float* __restrict__ b5t) {
    int tid = blockIdx.x * blockDim.x + threadIdx.x;
    if (tid >= TAB_ELEMS) return;
    int i = tid >> 8;     // IN index
    int o = tid & 255;    // OUT index

    float mu = mu_p[0];
    float sigma = sigma_p[0];
    float wv = w[o * IN_COUNT + i];
    float wn = (fminf(fmaxf(wv, 5.5f), 35.5f) - mu) / sigma;
    float g = softplus_precise(raw_gamma[o * IN_COUNT + i]) * (1.0f / (float)OUT_COUNT);

    float b[N_SPLINES];
#pragma unroll
    for (int s = 0; s < N_SPLINES; ++s) {
        const float* br = breaks + s * 17;
        float wc = fminf(fmaxf(wn, br[0]), br[16] - 1e-6f);
        // searchsorted(side='right') - 1, branch-free
        int idx = 0;
#pragma unroll
        for (int k = 1; k < 16; ++k) idx += (wc >= br[k]) ? 1 : 0;
        float t = wc - br[idx];
        const float* a = coefs + (s * 16 + idx) * 4;
        b[s] = ((a[0] * t + a[1]) * t + a[2]) * t + a[3];
    }

    float4 t4;
    t4.x = b[0] * g * LN2_F;
    t4.y = b[1] * LN2_F;
    t4.z = b[2] * LOG2E_F;
    t4.w = b[3];
    tab4[i * OUT_COUNT + o] = t4;
    b5t [i * OUT_COUNT + o] = b[4] * g;
}

// ---------------------------------------------------------------------------
// Kernel 2: fused main kernel. One wave32 owns a 16x16 (batch x out) C tile.
//   * linear term  Σ_i b5g[o,i]*x[b,i]  -> V_WMMA_F32_16X16X4_F32, K=4 chunks
//   * nonlinear term in native exp2/log2 space: 5 trans + ~6 VALU per element
// 4 waves/block share the relu(x) 16x256 tile in LDS (16 KB).
// ---------------------------------------------------------------------------
__global__ __launch_bounds__(128)
void kan_main_kernel(const float* __restrict__ x,
                     const float4* __restrict__ tab4,
                     const float* __restrict__ b5t,
                     float* __restrict__ out) {
    __shared__ float xs[16 * IN_COUNT];   // relu(x) tile, [m][i]

    const int b0 = blockIdx.x * 16;

    for (int t = threadIdx.x; t < 16 * IN_COUNT; t += 128) {
        xs[t] = fmaxf(x[b0 * IN_COUNT + t], 0.0f);
    }
    __syncthreads();

    const int wave   = threadIdx.x >> 5;
    const int lane   = threadIdx.x & 31;
    const int o_base = (blockIdx.y * 4 + wave) * 16;

    const int n        = lane & 15;      // N column (B/C/D role)
    const int half     = lane >> 4;      // 0: lanes 0-15, 1: lanes 16-31
    const int kb       = half * 2;       // A/B K offset for this half-wave
    const int row_base = half * 8;       // C-layout M rows
    const int oc       = o_base + n;     // this lane's output column

    v8f c = {};               // WMMA accumulator (linear b5 term)
    float acc[8];
#pragma unroll
    for (int r = 0; r < 8; ++r) acc[r] = 0.0f;

    for (int i0 = 0; i0 < IN_COUNT; i0 += 4) {
        // A: 16x4 f32 tile of relu(x). Lane<16: {K0,K1}; lane>=16: {K2,K3}.
        v2f a;
        a.x = xs[n * IN_COUNT + i0 + kb];
        a.y = xs[n * IN_COUNT + i0 + kb + 1];
        // B: 4x16 f32 tile of gamma-scaled b5, N striped across lanes.
        v2f b;
        b.x = b5t[(i0 + kb)     * OUT_COUNT + oc];
        b.y = b5t[(i0 + kb + 1) * OUT_COUNT + oc];
        c = __builtin_amdgcn_wmma_f32_16x16x4_f32(
                false, a, false, b, (short)0, c, false, false);

#pragma unroll
        for (int j = 0; j < 4; ++j) {
            const int i = i0 + j;
            const float4 s = tab4[i * OUT_COUNT + oc];  // one b128 load
#pragma unroll
            for (int r = 0; r < 8; ++r) {
                float xv   = xs[(row_base + r) * IN_COUNT + i];
                // base = expm1(b3*x)           (>= 0 since x>=0, b3>0)
                float base = fast_exp2(s.z * xv) - 1.0f;
                // p = base^b4 = exp2(b4*log2(base)); base==0 -> -inf -> 0 OK
                float p    = fast_exp2(s.w * fast_log2(base));
                // inner = b2*log1p(p); ln2 pre-folded into s.y
                float u    = s.y * fast_log2(1.0f + p);
                // term = log1p(u); ln2 & gamma/OUT pre-folded into s.x
                float t    = fast_log2(1.0f + u);
                acc[r] = fmaf(s.x, t, acc[r]);
            }
        }
    }

    // Combine linear (WMMA C) + nonlinear accumulators; fast softplus.
#pragma unroll
    for (int v = 0; v < 8; ++v) {
        float y  = c[v] + acc[v];
        float sp = fmaxf(y, 0.0f) +
                   LN2_F * fast_log2(1.0f + fast_exp2(-fabsf(y) * LOG2E_F));
        out[(b0 + row_base + v) * OUT_COUNT + oc] = sp;
    }
}

// ---------------------------------------------------------------------------
// Host-side launcher
// inputs: 0=x 1=w 2=raw_gamma 3=breaks 4=coefs 5=mu 6=sigma
// ---------------------------------------------------------------------------
extern "C" void kernel_launch(void* const* d_in, const int* in_sizes, int n_in,
                              void* d_out, int out_size, void* d_ws, size_t ws_size,
                              hipStream_t stream) {
    const float* x         = (const float*)d_in[0];
    const float* w         = (const float*)d_in[1];
    const float* raw_gamma = (const float*)d_in[2];
    const float* breaks    = (const float*)d_in[3];
    const float* coefs     = (const float*)d_in[4];
    const float* mu_p      = (const float*)d_in[5];
    const float* sigma_p   = (const float*)d_in[6];

    float4* tab4 = (float4*)d_ws;                         // 1.0 MB
    float*  b5t  = (float*)((char*)d_ws + TAB_ELEMS * sizeof(float4)); // +256 KB
    float*  out  = (float*)d_out;

    kan_spline_precompute<<<TAB_ELEMS / 256, 256, 0, stream>>>(
        w, raw_gamma, breaks, coefs, mu_p, sigma_p, tab4, b5t);

    dim3 grid(BATCH / 16, OUT_COUNT / 64);  // 128 x 4 blocks
    kan_main_kernel<<<grid, 128, 0, stream>>>(x, tab4, b5t, out);
}